// GCNs_50027779064033
// MI455X (gfx1250) — compile-verified
//
#include <hip/hip_runtime.h>

// ---------------------------------------------------------------------------
// GCN forward, 4 layers of:  h = relu( (h@W + b) scatter-summed over edges
//                                      + (h@W + b) )            [self-loop]
// MI455X / gfx1250: fp32 WMMA (V_WMMA_F32_16X16X4_F32) for the dense GEMM,
// L2-resident gather + global_atomic_add_f32 scatter for the edge phase.
// ---------------------------------------------------------------------------

typedef float v2f __attribute__((ext_vector_type(2)));
typedef float v8f __attribute__((ext_vector_type(8)));

#define N_NODES  10000
#define N_EDGES  640000
#define DIM      128
#define N_LAYERS 4

// ---------------------------------------------------------------------------
// GEMM + bias:  Y[row0:row0+16][:] = X @ W + b   (Y written to hl AND agg)
// One wave per 16x16 tile; 8 waves per block cover all 128 output columns
// for one 16-row strip. K-loop: 32 x wmma_f32_16x16x4_f32.
//
// fp32 WMMA operand layouts (CDNA5 ISA 7.12.2):
//   A (16x4):  lane = m + 16*(k>=2), vgpr = k&1
//   B (4x16):  lane = n + 16*(k>=2), vgpr = k&1
//   C/D(16x16): vgpr v, lanes 0-15 -> M=v, lanes 16-31 -> M=v+8; N = lane&15
// ---------------------------------------------------------------------------
__global__ __launch_bounds__(256) void gcn_gemm_bias(
    const float* __restrict__ X,     // [N_NODES][DIM]  input features
    const float* __restrict__ W,     // [DIM][DIM]      layer weight
    const float* __restrict__ bias,  // [DIM]           layer bias
    float* __restrict__ Yhl,         // [N_NODES][DIM]  hl output
    float* __restrict__ Yagg)        // [N_NODES][DIM]  agg init (= hl, self loop)
{
    const int wave  = threadIdx.x >> 5;        // 0..7  -> column tile
    const int lane  = threadIdx.x & 31;
    const int row0  = blockIdx.x * 16;         // 625 blocks * 16 = 10000 exact
    const int col0  = wave * 16;
    const int m     = lane & 15;               // row (A) / col (B) within tile
    const int khalf = (lane >> 4) * 2;         // 0 or 2: K offset per lane half

    const float* xrow = X + (size_t)(row0 + m) * DIM;

    v8f c = {};
    #pragma unroll 4
    for (int k0 = 0; k0 < DIM; k0 += 4) {
        v2f a, b;
        a.x = xrow[k0 + khalf + 0];
        a.y = xrow[k0 + khalf + 1];
        b.x = W[(size_t)(k0 + khalf + 0) * DIM + col0 + m];
        b.y = W[(size_t)(k0 + khalf + 1) * DIM + col0 + m];
        // 8 args: (neg_a, A, neg_b, B, c_mod, C, reuse_a, reuse_b)
        c = __builtin_amdgcn_wmma_f32_16x16x4_f32(
            false, a, false, b, (short)0, c, false, false);
    }

    const float bcol  = bias[col0 + m];
    const int   mbase = row0 + ((lane >> 4) << 3);   // +8 for upper lane half
    #pragma unroll
    for (int v = 0; v < 8; ++v) {
        const float  val = c[v] + bcol;
        const size_t idx = (size_t)(mbase + v) * DIM + (col0 + m);
        Yhl[idx]  = val;
        Yagg[idx] = val;   // self-loop contribution, initializes the scatter sum
    }
}

// ---------------------------------------------------------------------------
// Edge scatter: agg[dst[e]] += hl[src[e]]  for all edges.
// One wave per edge (uniform src/dst -> scalar loads); each lane moves a
// float4 (4 columns) -> 4 global f32 atomics. hl/agg are L2-resident (5 MB).
// ---------------------------------------------------------------------------
__global__ __launch_bounds__(256) void gcn_edge_scatter(
    const float* __restrict__ hl,
    float*       __restrict__ agg,
    const int*   __restrict__ src,
    const int*   __restrict__ dst)
{
    const int wave = threadIdx.x >> 5;
    const int lane = threadIdx.x & 31;
    const int edge = blockIdx.x * 8 + wave;    // 80000 blocks * 8 = 640000 exact

    const int s = src[edge];
    const int d = dst[edge];

    const float4 v = *(const float4*)(hl + (size_t)s * DIM + lane * 4);
    float* p = agg + (size_t)d * DIM + lane * 4;
    atomicAdd(p + 0, v.x);
    atomicAdd(p + 1, v.y);
    atomicAdd(p + 2, v.z);
    atomicAdd(p + 3, v.w);
}

// ---------------------------------------------------------------------------
// ReLU: out = max(agg, 0), float4 vectorized.
// ---------------------------------------------------------------------------
__global__ __launch_bounds__(256) void gcn_relu(
    const float* __restrict__ in, float* __restrict__ out, int n4)
{
    const int i = blockIdx.x * blockDim.x + threadIdx.x;
    if (i < n4) {
        float4 v = ((const float4*)in)[i];
        v.x = fmaxf(v.x, 0.0f);
        v.y = fmaxf(v.y, 0.0f);
        v.z = fmaxf(v.z, 0.0f);
        v.w = fmaxf(v.w, 0.0f);
        ((float4*)out)[i] = v;
    }
}

// ---------------------------------------------------------------------------
// Driver. Inputs (setup_inputs order):
//   d_in[0] node_feats f32 [10000,128]
//   d_in[1] src        i32 [640000]
//   d_in[2] dst        i32 [640000]
//   d_in[3] Ws         f32 [4,128,128]
//   d_in[4] bs         f32 [4,128]
// d_out: f32 [10000,128]
// d_ws layout: hl (5.12MB) | agg (5.12MB) | h (5.12MB)
// ---------------------------------------------------------------------------
extern "C" void kernel_launch(void* const* d_in, const int* in_sizes, int n_in,
                              void* d_out, int out_size, void* d_ws, size_t ws_size,
                              hipStream_t stream) {
    (void)in_sizes; (void)n_in; (void)out_size; (void)ws_size;

    const float* node_feats = (const float*)d_in[0];
    const int*   src        = (const int*)  d_in[1];
    const int*   dst        = (const int*)  d_in[2];
    const float* Ws         = (const float*)d_in[3];
    const float* bs         = (const float*)d_in[4];
    float*       out        = (float*)d_out;

    const size_t FEAT_BYTES = (size_t)N_NODES * DIM * sizeof(float);
    char* ws = (char*)d_ws;
    float* hlbuf  = (float*)(ws);
    float* aggbuf = (float*)(ws + FEAT_BYTES);
    float* hbuf   = (float*)(ws + 2 * FEAT_BYTES);

    const int gemm_blocks    = N_NODES / 16;              // 625
    const int scatter_blocks = (N_EDGES * 32) / 256;      // 80000
    const int n4             = (N_NODES * DIM) / 4;       // 320000
    const int relu_blocks    = (n4 + 255) / 256;          // 1250

    for (int l = 0; l < N_LAYERS; ++l) {
        const float* hin = (l == 0) ? node_feats : hbuf;
        const float* W   = Ws + (size_t)l * DIM * DIM;
        const float* b   = bs + (size_t)l * DIM;
        float*       hout = (l == N_LAYERS - 1) ? out : hbuf;

        gcn_gemm_bias<<<gemm_blocks, 256, 0, stream>>>(hin, W, b, hlbuf, aggbuf);
        gcn_edge_scatter<<<scatter_blocks, 256, 0, stream>>>(hlbuf, aggbuf, src, dst);
        gcn_relu<<<relu_blocks, 256, 0, stream>>>(aggbuf, hout, n4);
    }
}